// DifferentiablePPO_65627100282923
// MI455X (gfx1250) — compile-verified
//
#include <hip/hip_runtime.h>
#include <math.h>

// ---------------------------------------------------------------------------
// PPO loss (GAE scan + normalized advantages + clipped surrogate/value loss)
// T=1024, N=8192.  Memory-bound streaming pipeline, 5 kernels on one stream.
// CDNA5 paths: global_prefetch_b8 (scan read-ahead), v_wmma_f32_16x16x4_f32
// (ones-matmul cross-lane reduction), global_atomic_add_f64.
// ---------------------------------------------------------------------------

#define T_STEPS 1024
#define N_ENVS  8192
#define CNT     (T_STEPS * N_ENVS)

static constexpr float GAMMA_F   = 0.99f;
static constexpr float LAMBDA_F  = 0.95f;
static constexpr float CLIP_F    = 0.2f;

typedef float v2f __attribute__((ext_vector_type(2)));
typedef float v8f __attribute__((ext_vector_type(8)));

// --- full 32-lane sum via two chained f32 WMMAs with an all-ones B matrix ---
// A-layout (16x4 f32): lane L<16 -> (M=L,K=0)/(M=L,K=1); lane L>=16 -> K=2/K=3.
// With a = {q, 0}: D[m][n] = q[m] + q[m+16]  (replicated over n).
// Folding D's 8 VGPRs into a second A and multiplying by ones again yields
// the total sum broadcast to every element of D2.  EXEC must be all ones.
__device__ __forceinline__ float wmma_full_sum32(float q) {
  v2f a;    a.x = q;    a.y = 0.0f;
  v2f ones; ones.x = 1.0f; ones.y = 1.0f;
  v8f c = {};
  v8f d = __builtin_amdgcn_wmma_f32_16x16x4_f32(
      false, a, false, ones, (short)0, c, false, false);
  v2f a2;
  a2.x = d[0] + d[2] + d[4] + d[6];
  a2.y = d[1] + d[3] + d[5] + d[7];
  v8f d2 = __builtin_amdgcn_wmma_f32_16x16x4_f32(
      false, a2, false, ones, (short)0, c, false, false);
  return d2[0];
}

// Block(256)-wide sum; result valid in threads 0..31 (used by thread 0).
__device__ __forceinline__ float block_sum_256(float v, float* lds) {
  const int tid = threadIdx.x;
  lds[tid] = v;
  __syncthreads();
  float total = 0.0f;
  if (tid < 32) {                      // wave 0, EXEC all ones inside
    float q = 0.0f;
#pragma unroll
    for (int i = 0; i < 8; ++i) q += lds[tid + 32 * i];
    total = wmma_full_sum32(q);
  }
  __syncthreads();
  return total;
}

// ---------------------------------------------------------------------------
__global__ void ppo_init_accum(double* accum) {
#pragma unroll
  for (int i = 0; i < 5; ++i) accum[i] = 0.0;
}

// ---------------------------------------------------------------------------
// Pass A: reverse GAE scan.  One thread per environment; lanes of a wave read
// consecutive addresses at each t (fully coalesced).  Prefetch 16 steps ahead.
__global__ __launch_bounds__(256) void ppo_gae_scan(
    const float* __restrict__ rewards,      // (T,N)
    const float* __restrict__ masks,        // (T+1,N)
    const float* __restrict__ bad_masks,    // (T+1,N)
    const float* __restrict__ value_preds,  // (T+1,N)
    const float* __restrict__ last_value,   // (N)
    float* __restrict__ returns,            // (T,N) out (workspace)
    double* __restrict__ accum)             // [0]=sum(adv) [1]=sum(adv^2)
{
  __shared__ float lds[256];
  const int n = blockIdx.x * 256 + threadIdx.x;

  float v_next = last_value[n];            // vp[T] := last_value
  float gae = 0.0f, s1 = 0.0f, s2 = 0.0f;
  const float kGL = GAMMA_F * LAMBDA_F;
  const int PF = 16;

  for (int t = T_STEPS - 1; t >= 0; --t) {
    const int idx  = t * N_ENVS + n;
    const int idx1 = idx + N_ENVS;         // (t+1)*N + n

    const float r  = rewards[idx];
    const float m  = masks[idx1];
    const float bd = bad_masks[idx1];
    const float v  = value_preds[idx];

    if (t >= PF) {                         // read-ahead: global_prefetch_b8
      const int p = idx - PF * N_ENVS;
      __builtin_prefetch(&rewards[p], 0, 3);
      __builtin_prefetch(&value_preds[p], 0, 3);
      __builtin_prefetch(&masks[p + N_ENVS], 0, 3);
      __builtin_prefetch(&bad_masks[p + N_ENVS], 0, 3);
    }

    const float delta = fmaf(GAMMA_F * m, v_next, r) - v;
    gae = fmaf(kGL * m, gae, delta) * bd;
    returns[idx] = gae + v;                // returns_t
    s1 += gae;                             // advantage == gae
    s2 = fmaf(gae, gae, s2);
    v_next = v;
  }

  const float b1 = block_sum_256(s1, lds);
  const float b2 = block_sum_256(s2, lds);
  if (threadIdx.x == 0) {
    atomicAdd(&accum[0], (double)b1);
    atomicAdd(&accum[1], (double)b2);
  }
}

// ---------------------------------------------------------------------------
// Stats: mean and 1/(std(ddof=1)+1e-5) once.
__global__ void ppo_stats(const double* __restrict__ accum,
                          float* __restrict__ fstats) {
  const double cnt = (double)CNT;
  const double s1 = accum[0], s2 = accum[1];
  const double mean = s1 / cnt;
  const double var  = (s2 - s1 * s1 / cnt) / (cnt - 1.0);
  fstats[0] = (float)mean;
  fstats[1] = 1.0f / ((float)sqrt(var) + 1e-5f);
}

// ---------------------------------------------------------------------------
// Pass B: elementwise PPO math, float4-vectorized grid-stride, 3 reductions.
__global__ __launch_bounds__(256) void ppo_loss_sums(
    const float4* __restrict__ ret4,
    const float4* __restrict__ vp4,        // value_preds[:-1]
    const float4* __restrict__ lp4,
    const float4* __restrict__ plp4,
    const float4* __restrict__ nv4,
    const float4* __restrict__ ent4,
    const float* __restrict__ fstats,
    double* __restrict__ accum)            // [2]=sum(min) [3]=sum(vmax) [4]=sum(ent)
{
  __shared__ float lds[256];
  const float mean = fstats[0];
  const float inv  = fstats[1];

  float a_min = 0.0f, a_vm = 0.0f, a_e = 0.0f;
  const int nvec = CNT / 4;
  const int stride = gridDim.x * blockDim.x;

  for (int i = blockIdx.x * blockDim.x + threadIdx.x; i < nvec; i += stride) {
    const float4 rt = ret4[i];
    const float4 vp = vp4[i];
    const float4 lp = lp4[i];
    const float4 pl = plp4[i];
    const float4 nv = nv4[i];
    const float4 en = ent4[i];

    const float rts[4] = {rt.x, rt.y, rt.z, rt.w};
    const float vps[4] = {vp.x, vp.y, vp.z, vp.w};
    const float lps[4] = {lp.x, lp.y, lp.z, lp.w};
    const float pls[4] = {pl.x, pl.y, pl.z, pl.w};
    const float nvs[4] = {nv.x, nv.y, nv.z, nv.w};
    const float ens[4] = {en.x, en.y, en.z, en.w};

#pragma unroll
    for (int k = 0; k < 4; ++k) {
      const float adv   = (rts[k] - vps[k] - mean) * inv;     // normalized
      const float ratio = __expf(lps[k] - pls[k]);
      const float rcl   = fminf(fmaxf(ratio, 1.0f - CLIP_F), 1.0f + CLIP_F);
      a_min += fminf(ratio * adv, rcl * adv);

      const float dcl = fminf(fmaxf(nvs[k] - vps[k], -CLIP_F), CLIP_F);
      const float vpc = vps[k] + dcl;
      const float d1  = nvs[k] - rts[k];
      const float d2  = vpc    - rts[k];
      a_vm += fmaxf(d1 * d1, d2 * d2);
      a_e  += ens[k];
    }
  }

  const float bmn = block_sum_256(a_min, lds);
  const float bvm = block_sum_256(a_vm, lds);
  const float be  = block_sum_256(a_e,  lds);
  if (threadIdx.x == 0) {
    atomicAdd(&accum[2], (double)bmn);
    atomicAdd(&accum[3], (double)bvm);
    atomicAdd(&accum[4], (double)be);
  }
}

// ---------------------------------------------------------------------------
__global__ void ppo_finalize(const double* __restrict__ accum,
                             float* __restrict__ out) {
  const double cnt = (double)CNT;
  const double action_loss = -accum[2] / cnt;
  const double value_loss  = 0.5 * (accum[3] / cnt);
  const double ent_mean    = accum[4] / cnt;
  out[0] = (float)(value_loss * 0.5 + action_loss - ent_mean * 0.01);
}

// ---------------------------------------------------------------------------
extern "C" void kernel_launch(void* const* d_in, const int* in_sizes, int n_in,
                              void* d_out, int out_size, void* d_ws, size_t ws_size,
                              hipStream_t stream) {
  (void)in_sizes; (void)n_in; (void)out_size; (void)ws_size;

  const float* rewards     = (const float*)d_in[0];  // (T,N,1)
  const float* masks       = (const float*)d_in[1];  // (T+1,N,1)
  const float* bad_masks   = (const float*)d_in[2];  // (T+1,N,1)
  const float* value_preds = (const float*)d_in[3];  // (T+1,N,1)
  const float* last_value  = (const float*)d_in[4];  // (N,1)
  const float* log_prob    = (const float*)d_in[5];  // (T,N,1)
  const float* prev_lp     = (const float*)d_in[6];  // (T,N,1)
  const float* new_value   = (const float*)d_in[7];  // (T,N,1)
  const float* dist_ent    = (const float*)d_in[8];  // (T,N,1)

  // workspace: [0..39] 5 doubles, [64..71] 2 floats, [256..] returns (32 MB)
  double* accum  = (double*)d_ws;
  float*  fstats = (float*)((char*)d_ws + 64);
  float*  returns = (float*)((char*)d_ws + 256);

  ppo_init_accum<<<1, 1, 0, stream>>>(accum);

  ppo_gae_scan<<<N_ENVS / 256, 256, 0, stream>>>(
      rewards, masks, bad_masks, value_preds, last_value, returns, accum);

  ppo_stats<<<1, 1, 0, stream>>>(accum, fstats);

  ppo_loss_sums<<<1024, 256, 0, stream>>>(
      (const float4*)returns, (const float4*)value_preds,
      (const float4*)log_prob, (const float4*)prev_lp,
      (const float4*)new_value, (const float4*)dist_ent,
      fstats, accum);

  ppo_finalize<<<1, 1, 0, stream>>>(accum, (float*)d_out);
}